// HSNLayer_19061064860111
// MI455X (gfx1250) — compile-verified
//
#include <hip/hip_runtime.h>
#include <cmath>

typedef __attribute__((ext_vector_type(2))) float v2f;
typedef __attribute__((ext_vector_type(8))) float v8f;

#define PAIR_PITCH 272  // float2 units; 2*272 % 64 == 32 -> half-waves hit disjoint bank halves
#define KCHUNK      32  // K rows of W per LDS stage -> 16 pair-rows * 272 * 8B = 34816 B

// ---------------------------------------------------------------------------
// Y[M,256] = X[M,256] @ W[256,256], exact f32 via V_WMMA_F32_16X16X4_F32.
// Workgroup: 256 threads = 8 waves as 2(M) x 4(N); WG tile = 64 x 256.
// Wave tile: 32(M) x 64(N) -> acc[2][4] of v8f.
// W staged in LDS K-pair-interleaved: lds2[p*PITCH+n] = {W[2p][n], W[2p+1][n]}
// so each B fragment is one aligned ds_load_b64 (no repacking movs).
// ---------------------------------------------------------------------------
__global__ __launch_bounds__(256)
void gemm256_wmma(const float* __restrict__ X, const float* __restrict__ W,
                  float* __restrict__ Y, int M)
{
    __shared__ v2f ldsW2[(KCHUNK / 2) * PAIR_PITCH];

    const int t    = threadIdx.x;
    const int wave = t >> 5;
    const int lane = t & 31;
    const int half = lane >> 4;   // 0: K+{0,1}, 1: K+{2,3} per 16x4 f32 A layout
    const int lm   = lane & 15;
    const int wm   = wave >> 2;   // 0..1  (M within WG)
    const int wn   = wave & 3;    // 0..3  (N within WG)

    const int mbase = blockIdx.x * 64 + wm * 32;
    const int nbase = wn * 64;

    int r0 = mbase + lm;       if (r0 >= M) r0 = M - 1;   // clamp (dup rows harmless)
    int r1 = mbase + 16 + lm;  if (r1 >= M) r1 = M - 1;
    const float* x0 = X + (size_t)r0 * 256;
    const float* x1 = X + (size_t)r1 * 256;

    v8f acc[2][4];
#pragma unroll
    for (int i = 0; i < 2; ++i)
#pragma unroll
        for (int j = 0; j < 4; ++j)
#pragma unroll
            for (int v = 0; v < 8; ++v) acc[i][j][v] = 0.0f;

    for (int kc = 0; kc < 256; kc += KCHUNK) {
        __syncthreads();   // previous chunk fully consumed before overwrite
        // stage W[kc..kc+31][:] pair-interleaved: 16 pair-rows x 256 cols
#pragma unroll
        for (int i = 0; i < 4; ++i) {
            const int g = i * 256 + t;        // 0..1023 four-column groups
            const int p = g >> 6;             // pair row 0..15
            const int c = (g & 63) * 4;       // column 0..252
            const float4 e0 = *reinterpret_cast<const float4*>(
                W + (size_t)(kc + 2 * p) * 256 + c);
            const float4 e1 = *reinterpret_cast<const float4*>(
                W + (size_t)(kc + 2 * p + 1) * 256 + c);
            v2f* dst = &ldsW2[p * PAIR_PITCH + c];
            v2f q0; q0.x = e0.x; q0.y = e1.x;
            v2f q1; q1.x = e0.y; q1.y = e1.y;
            v2f q2; q2.x = e0.z; q2.y = e1.z;
            v2f q3; q3.x = e0.w; q3.y = e1.w;
            dst[0] = q0; dst[1] = q1; dst[2] = q2; dst[3] = q3;
        }
        __syncthreads();

#pragma unroll
        for (int k = 0; k < KCHUNK; k += 4) {
            const int kg = kc + k + half * 2;            // global K for this half-wave
            const int p  = (k >> 1) + half;              // local pair row
            const v2f a0 = *reinterpret_cast<const v2f*>(x0 + kg);  // {K, K+1}
            const v2f a1 = *reinterpret_cast<const v2f*>(x1 + kg);
            const v2f* brow = &ldsW2[p * PAIR_PITCH + nbase + lm];
#pragma unroll
            for (int j = 0; j < 4; ++j) {
                const v2f b = brow[j * 16];              // one ds_load_b64
                acc[0][j] = __builtin_amdgcn_wmma_f32_16x16x4_f32(
                    false, a0, false, b, (short)0, acc[0][j], false, false);
                acc[1][j] = __builtin_amdgcn_wmma_f32_16x16x4_f32(
                    false, a1, false, b, (short)0, acc[1][j], false, false);
            }
        }
    }

    // D layout: VGPR v -> row v (lanes 0-15) / row v+8 (lanes 16-31), col = lane&15
#pragma unroll
    for (int i = 0; i < 2; ++i) {
#pragma unroll
        for (int v = 0; v < 8; ++v) {
            const int row = mbase + i * 16 + half * 8 + v;
            if (row < M) {
                float* yr = Y + (size_t)row * 256 + nbase + lm;
#pragma unroll
                for (int j = 0; j < 4; ++j) yr[j * 16] = acc[i][j][v];
            }
        }
    }
}

// ---------------------------------------------------------------------------
// Y[rows[e], :] += vals[e] * Xw[cols[e], :]   (CH = 256), one wave per nnz.
// HW f32 atomics (no return) -> global_atomic_add_f32 in L2.
// ---------------------------------------------------------------------------
__device__ __forceinline__ void fadd_atomic(float* p, float v)
{
    __hip_atomic_fetch_add(p, v, __ATOMIC_RELAXED, __HIP_MEMORY_SCOPE_AGENT);
}

__global__ __launch_bounds__(256)
void spmm_scatter(const int* __restrict__ rows, const int* __restrict__ cols,
                  const float* __restrict__ vals, const float* __restrict__ Xw,
                  float* __restrict__ Y, int nnz)
{
    const int wave = threadIdx.x >> 5;
    const int lane = threadIdx.x & 31;
    const int e = blockIdx.x * 8 + wave;
    if (e >= nnz) return;

    const int   r = rows[e];
    const int   c = cols[e];
    const float v = vals[e];
    const float* src = Xw + (size_t)c * 256;
    float*       dst = Y  + (size_t)r * 256;

#pragma unroll
    for (int h = 0; h < 2; ++h) {
        const int base = h * 128 + lane * 4;
        const float4 d = *reinterpret_cast<const float4*>(src + base);
        fadd_atomic(dst + base + 0, v * d.x);
        fadd_atomic(dst + base + 1, v * d.y);
        fadd_atomic(dst + base + 2, v * d.z);
        fadd_atomic(dst + base + 3, v * d.w);
    }
}

__global__ __launch_bounds__(256)
void zero4(float4* __restrict__ p, size_t n4)
{
    const size_t i = (size_t)blockIdx.x * 256 + threadIdx.x;
    if (i < n4) p[i] = make_float4(0.f, 0.f, 0.f, 0.f);
}

__global__ __launch_bounds__(256)
void sigmoid4(float4* __restrict__ p, size_t n4)
{
    const size_t i = (size_t)blockIdx.x * 256 + threadIdx.x;
    if (i < n4) {
        float4 d = p[i];
        d.x = 1.0f / (1.0f + __expf(-d.x));
        d.y = 1.0f / (1.0f + __expf(-d.y));
        d.z = 1.0f / (1.0f + __expf(-d.z));
        d.w = 1.0f / (1.0f + __expf(-d.w));
        p[i] = d;
    }
}

// ---------------------------------------------------------------------------
extern "C" void kernel_launch(void* const* d_in, const int* in_sizes, int n_in,
                              void* d_out, int out_size, void* d_ws, size_t ws_size,
                              hipStream_t stream)
{
    (void)in_sizes; (void)n_in; (void)out_size; (void)ws_size;

    const float* x        = (const float*)d_in[0];
    const int*   adj_rows = (const int*)  d_in[1];
    const int*   adj_cols = (const int*)  d_in[2];
    const float* adj_vals = (const float*)d_in[3];
    const int*   inc_rows = (const int*)  d_in[4];
    const int*   inc_cols = (const int*)  d_in[5];
    const float* inc_vals = (const float*)d_in[6];
    const float* W00      = (const float*)d_in[7];
    const float* W01      = (const float*)d_in[8];
    const float* W10      = (const float*)d_in[9];
    const float* W11      = (const float*)d_in[10];
    float*       out      = (float*)d_out;

    const int    N_NODES = 50000, N_EDGES = 400000, NNZ = 800000;
    const size_t NE = (size_t)N_NODES * 256;   // 12.8M f32
    const size_t EE = (size_t)N_EDGES * 256;   // 102.4M f32

    float* T1 = (float*)d_ws;      // xw0, later xn2
    float* T2 = T1 + NE;           // xw1
    float* T3 = T2 + NE;           // x_nodes_l1
    float* E1 = T3 + NE;           // x_edges_l1
    float* E2 = E1 + EE;           // xe2

    const dim3 blk(256);
    const int  gN  = (N_NODES + 63) / 64;          // 782
    const int  gE  = (N_EDGES + 63) / 64;          // 6250
    const int  gS  = (NNZ + 7) / 8;                // 100000
    const int  gZN = (int)((NE / 4 + 255) / 256);
    const int  gZE = (int)((EE / 4 + 255) / 256);

    // Level 1 GEMMs
    gemm256_wmma<<<dim3(gN), blk, 0, stream>>>(x, W00, T1, N_NODES);
    gemm256_wmma<<<dim3(gN), blk, 0, stream>>>(x, W01, T2, N_NODES);

    // x_nodes_l1 = sigmoid(A0 @ xw0)
    zero4<<<dim3(gZN), blk, 0, stream>>>((float4*)T3, NE / 4);
    spmm_scatter<<<dim3(gS), blk, 0, stream>>>(adj_rows, adj_cols, adj_vals, T1, T3, NNZ);
    sigmoid4<<<dim3(gZN), blk, 0, stream>>>((float4*)T3, NE / 4);

    // x_edges_l1 = sigmoid(B1^T @ xw1)  (rows = inc_cols edge ids, cols = inc_rows node ids)
    zero4<<<dim3(gZE), blk, 0, stream>>>((float4*)E1, EE / 4);
    spmm_scatter<<<dim3(gS), blk, 0, stream>>>(inc_cols, inc_rows, inc_vals, T2, E1, NNZ);
    sigmoid4<<<dim3(gZE), blk, 0, stream>>>((float4*)E1, EE / 4);

    // Level 2 GEMMs
    gemm256_wmma<<<dim3(gN), blk, 0, stream>>>(T3, W10, T1, N_NODES);   // reuse T1
    gemm256_wmma<<<dim3(gE), blk, 0, stream>>>(E1, W11, E2, N_EDGES);

    // out = sigmoid(A0 @ xn2 + B1 @ xe2)
    zero4<<<dim3(gZN), blk, 0, stream>>>((float4*)out, NE / 4);
    spmm_scatter<<<dim3(gS), blk, 0, stream>>>(adj_rows, adj_cols, adj_vals, T1, out, NNZ);
    spmm_scatter<<<dim3(gS), blk, 0, stream>>>(inc_rows, inc_cols, inc_vals, E2, out, NNZ);
    sigmoid4<<<dim3(gZN), blk, 0, stream>>>((float4*)out, NE / 4);
}